// CosGNN_14894946583455
// MI455X (gfx1250) — compile-verified
//
#include <hip/hip_runtime.h>
#include <math.h>

#define N_NODES 100000
#define N_EDGES 1600000
#define IN_F    128
#define HD      128      // H*D = 2*64
#define NEG_SLOPE 0.2f

typedef float v2f __attribute__((ext_vector_type(2)));
typedef float v8f __attribute__((ext_vector_type(8)));
typedef int   v4i __attribute__((ext_vector_type(4)));

#if __has_builtin(__builtin_amdgcn_global_load_async_to_lds_b128) && \
    __has_builtin(__builtin_amdgcn_s_wait_asynccnt)
#define HAVE_ASYNC_LDS 1
typedef __attribute__((address_space(1))) v4i* gvec_t;   // global int4*
typedef __attribute__((address_space(3))) v4i* lvec_t;   // LDS int4*
#endif

// ---- monotone float <-> uint encoding for atomicMax on floats ----
__device__ __forceinline__ unsigned enc_f32(float f) {
    unsigned b = __float_as_uint(f);
    return (b & 0x80000000u) ? ~b : (b | 0x80000000u);
}
__device__ __forceinline__ float dec_f32(unsigned u) {
    return __uint_as_float((u & 0x80000000u) ? (u ^ 0x80000000u) : ~u);
}

// =====================================================================
// Kernel 1: node projections via fp32 WMMA.
//   fc_src = feat @ W_src.T            -> workspace
//   out    = feat @ W_dst.T + b_dst    -> d_out (aggregation adds on top)
// Block: 128 threads = 4 waves, handles a 16-node tile.
// Each wave owns 4 of the 16 (matrix, n-tile) output tiles.
// Feat tile staged via GLOBAL_LOAD_ASYNC_TO_LDS (ASYNCcnt) when available.
// =====================================================================
__launch_bounds__(128)
__global__ void gat_node_gemm(const float* __restrict__ feat,
                              const float* __restrict__ Wsrc,
                              const float* __restrict__ Wdst,
                              const float* __restrict__ bdst,
                              float* __restrict__ fc_src,
                              float* __restrict__ out) {
    __shared__ float tileA[16][IN_F + 4];   // pad 4 floats -> conflict-free column reads
    const int node0 = blockIdx.x * 16;
    const int tid   = threadIdx.x;

    // cooperative load: 16x128 f32 tile, 16B per thread per iteration
#pragma unroll
    for (int i = 0; i < 4; ++i) {
        int f   = tid + i * 128;            // float4 index in [0,512)
        int row = f >> 5;
        int c4  = f & 31;
        const float* gsrc = feat + (size_t)(node0 + row) * IN_F + c4 * 4;
#ifdef HAVE_ASYNC_LDS
        __builtin_amdgcn_global_load_async_to_lds_b128(
            (gvec_t)const_cast<float*>(gsrc),
            (lvec_t)&tileA[row][c4 * 4], 0, 0);
#else
        *(float4*)&tileA[row][c4 * 4] = *(const float4*)gsrc;
#endif
    }
#ifdef HAVE_ASYNC_LDS
    __builtin_amdgcn_s_wait_asynccnt(0);
#endif
    __syncthreads();

    const int lane  = tid & 31;
    const int wave  = tid >> 5;
    const int mrow  = lane & 15;            // M (A) / N (B) index
    const int khi   = (lane >> 4) << 1;     // K sub-offset: 0 (lanes 0-15) or 2 (16-31)
    const int rbase = (lane >> 4) << 3;     // D-matrix row base: 0 or 8

    // 4 tiles per wave: t in [wave*4, wave*4+4); t<8 -> W_src, t>=8 -> W_dst
    const float* wrow[4];
    int n0[4];
#pragma unroll
    for (int i = 0; i < 4; ++i) {
        int t = wave * 4 + i;
        n0[i] = (t & 7) * 16;
        const float* W = (t < 8) ? Wsrc : Wdst;
        wrow[i] = W + (size_t)(n0[i] + mrow) * IN_F;   // row of W == column of W.T
    }

    v8f acc[4] = {};
    for (int k = 0; k < IN_F; k += 4) {
        int kb = k + khi;
        v2f a = *(const v2f*)&tileA[mrow][kb];          // A frag shared by 4 WMMAs
#pragma unroll
        for (int i = 0; i < 4; ++i) {
            v2f b;
            b.x = wrow[i][kb];
            b.y = wrow[i][kb + 1];
            acc[i] = __builtin_amdgcn_wmma_f32_16x16x4_f32(
                false, a, false, b, (short)0, acc[i], false, false);
        }
    }

#pragma unroll
    for (int i = 0; i < 4; ++i) {
        int t = wave * 4 + i;
        float bias = (t >= 8) ? bdst[n0[i] + mrow] : 0.0f;
#pragma unroll
        for (int v = 0; v < 8; ++v) {
            int node = node0 + v + rbase;               // D: M = v + rbase, N = mrow
            float val = acc[i][v] + bias;
            if (t < 8) fc_src[(size_t)node * HD + n0[i] + mrow] = val;
            else       out   [(size_t)node * HD + n0[i] + mrow] = val;
        }
    }
}

// =====================================================================
// Kernel 2: attn_src/attn_dst = feat @ a_{src,dst}.T   [N,2] each.
// One wave per node; float4 partials + shuffle reduction.
// =====================================================================
__launch_bounds__(256)
__global__ void gat_attn(const float* __restrict__ feat,
                         const float* __restrict__ a_src,
                         const float* __restrict__ a_dst,
                         float* __restrict__ attn_s,
                         float* __restrict__ attn_d) {
    int gw   = (blockIdx.x * blockDim.x + threadIdx.x) >> 5;
    int lane = threadIdx.x & 31;
    if (gw >= N_NODES) return;

    float4 f   = ((const float4*)(feat + (size_t)gw * IN_F))[lane];
    float4 s0v = ((const float4*)(a_src))[lane];
    float4 s1v = ((const float4*)(a_src + IN_F))[lane];
    float4 d0v = ((const float4*)(a_dst))[lane];
    float4 d1v = ((const float4*)(a_dst + IN_F))[lane];

    float p0 = f.x*s0v.x + f.y*s0v.y + f.z*s0v.z + f.w*s0v.w;
    float p1 = f.x*s1v.x + f.y*s1v.y + f.z*s1v.z + f.w*s1v.w;
    float p2 = f.x*d0v.x + f.y*d0v.y + f.z*d0v.z + f.w*d0v.w;
    float p3 = f.x*d1v.x + f.y*d1v.y + f.z*d1v.z + f.w*d1v.w;
#pragma unroll
    for (int off = 16; off > 0; off >>= 1) {
        p0 += __shfl_xor(p0, off);
        p1 += __shfl_xor(p1, off);
        p2 += __shfl_xor(p2, off);
        p3 += __shfl_xor(p3, off);
    }
    if (lane == 0) {
        attn_s[gw * 2 + 0] = p0;
        attn_s[gw * 2 + 1] = p1;
        attn_d[gw * 2 + 0] = p2;
        attn_d[gw * 2 + 1] = p3;
    }
}

// =====================================================================
// Kernel 3: init per-node softmax state (max encoding + denom).
// =====================================================================
__global__ void gat_init(unsigned* __restrict__ m_u, float* __restrict__ denom) {
    int i = blockIdx.x * blockDim.x + threadIdx.x;
    if (i < 2 * N_NODES) { m_u[i] = 0u; denom[i] = 0.0f; }   // enc(x) > 0 for all x
}

__device__ __forceinline__ float edge_logit(const float* as, const float* ad,
                                            int s, int d, int h) {
    float ev = as[s * 2 + h] + ad[d * 2 + h];
    return ev > 0.0f ? ev : NEG_SLOPE * ev;
}

// =====================================================================
// Kernel 4: segment max over dst (monotone-uint atomicMax).
// =====================================================================
__global__ void gat_edge_max(const int* __restrict__ src, const int* __restrict__ dst,
                             const float* __restrict__ as, const float* __restrict__ ad,
                             unsigned* __restrict__ m_u) {
    int e = blockIdx.x * blockDim.x + threadIdx.x;
    if (e >= N_EDGES) return;
    int s = src[e], d = dst[e];
#pragma unroll
    for (int h = 0; h < 2; ++h)
        atomicMax(&m_u[d * 2 + h], enc_f32(edge_logit(as, ad, s, d, h)));
}

// =====================================================================
// Kernel 5: denom = segment_sum(exp(e - m[dst])).
// =====================================================================
__global__ void gat_edge_den(const int* __restrict__ src, const int* __restrict__ dst,
                             const float* __restrict__ as, const float* __restrict__ ad,
                             const unsigned* __restrict__ m_u, float* __restrict__ denom) {
    int e = blockIdx.x * blockDim.x + threadIdx.x;
    if (e >= N_EDGES) return;
    int s = src[e], d = dst[e];
#pragma unroll
    for (int h = 0; h < 2; ++h) {
        float ex = __expf(edge_logit(as, ad, s, d, h) - dec_f32(m_u[d * 2 + h]));
        atomicAdd(&denom[d * 2 + h], ex);
    }
}

// =====================================================================
// Kernel 6: wave-per-edge aggregation.
// lane l owns features [4l,4l+4); head = l>>4. out += a * fc_src[src].
// =====================================================================
__launch_bounds__(256)
__global__ void gat_edge_agg(const int* __restrict__ src, const int* __restrict__ dst,
                             const float* __restrict__ as, const float* __restrict__ ad,
                             const unsigned* __restrict__ m_u, const float* __restrict__ denom,
                             const float* __restrict__ fc_src, float* __restrict__ out) {
    int gw   = (blockIdx.x * blockDim.x + threadIdx.x) >> 5;
    int lane = threadIdx.x & 31;
    if (gw >= N_EDGES) return;
    int s = src[gw], d = dst[gw];
    int h = lane >> 4;

    float ex   = __expf(edge_logit(as, ad, s, d, h) - dec_f32(m_u[d * 2 + h]));
    float coef = ex / fmaxf(denom[d * 2 + h], 1e-38f);

    float4 v = ((const float4*)(fc_src + (size_t)s * HD))[lane];
    float* o = out + (size_t)d * HD + lane * 4;
    atomicAdd(o + 0, v.x * coef);
    atomicAdd(o + 1, v.y * coef);
    atomicAdd(o + 2, v.z * coef);
    atomicAdd(o + 3, v.w * coef);
}

// =====================================================================
extern "C" void kernel_launch(void* const* d_in, const int* in_sizes, int n_in,
                              void* d_out, int out_size, void* d_ws, size_t ws_size,
                              hipStream_t stream) {
    const float* feat  = (const float*)d_in[0];
    const int*   src   = (const int*)  d_in[1];
    const int*   dst   = (const int*)  d_in[2];
    const float* Wsrc  = (const float*)d_in[3];
    const float* Wdst  = (const float*)d_in[4];
    const float* bdst  = (const float*)d_in[5];
    const float* a_src = (const float*)d_in[6];
    const float* a_dst = (const float*)d_in[7];
    float* out = (float*)d_out;

    // workspace layout (256B aligned blocks)
    char* ws = (char*)d_ws;
    size_t off = 0;
    auto alloc = [&](size_t bytes) { void* p = ws + off;
        off = (off + bytes + 255) & ~(size_t)255; return p; };
    float*    fc_src = (float*)   alloc((size_t)N_NODES * HD * sizeof(float)); // 51.2 MB
    float*    attn_s = (float*)   alloc((size_t)N_NODES * 2 * sizeof(float));
    float*    attn_d = (float*)   alloc((size_t)N_NODES * 2 * sizeof(float));
    unsigned* m_u    = (unsigned*)alloc((size_t)N_NODES * 2 * sizeof(unsigned));
    float*    denom  = (float*)   alloc((size_t)N_NODES * 2 * sizeof(float));
    (void)ws_size; (void)in_sizes; (void)n_in; (void)out_size;

    gat_node_gemm<<<N_NODES / 16, 128, 0, stream>>>(feat, Wsrc, Wdst, bdst, fc_src, out);
    gat_attn    <<<(N_NODES + 7) / 8, 256, 0, stream>>>(feat, a_src, a_dst, attn_s, attn_d);
    gat_init    <<<(2 * N_NODES + 255) / 256, 256, 0, stream>>>(m_u, denom);
    gat_edge_max<<<(N_EDGES + 255) / 256, 256, 0, stream>>>(src, dst, attn_s, attn_d, m_u);
    gat_edge_den<<<(N_EDGES + 255) / 256, 256, 0, stream>>>(src, dst, attn_s, attn_d, m_u, denom);
    gat_edge_agg<<<(N_EDGES + 7) / 8, 256, 0, stream>>>(src, dst, attn_s, attn_d, m_u, denom, fc_src, out);
}